// hnet_structure_deconv_85143431676509
// MI455X (gfx1250) — compile-verified
//
#include <hip/hip_runtime.h>
#include <cstddef>

typedef __attribute__((ext_vector_type(2))) float v2f;
typedef __attribute__((ext_vector_type(4))) float v4f;
typedef __attribute__((ext_vector_type(8))) float v8f;

// Inverse of OUTPUT_ORDER: final output row for original row j.
__device__ __constant__ int d_inv_order[26] = {
    15,14,17,16,19,18,21,20,23,22,25,24,
     1, 0, 3, 2, 5, 4, 7, 6, 9, 8,11,10,13,12};

struct BranchParams {
    const float* W[6];    // ConvTranspose1d weight (in_c, out_c, 2)
    const float* b[6];    // conv bias (out_c)
    const float* g[6];    // BN gamma
    const float* bt[6];   // BN beta
    const float* mn[6];   // BN running mean
    const float* vr[6];   // BN running var
};

// ---------------------------------------------------------------------------
// Phase A: per-(branch,row) deconv+BN+ReLU pipeline, entirely in LDS.
// grid.x = nb (one workgroup per bit-pattern row), 256 threads.
// Writes flattened activation (c-major, then l) of length K to act[m*K ..].
// ---------------------------------------------------------------------------
__global__ __launch_bounds__(256)
void hnet_branch_forward(BranchParams p, const float* __restrict__ sow,
                         float* __restrict__ act, int c_in, int n_layers) {
    extern __shared__ float smem[];
    float* buf0 = smem;            // 12288 floats
    float* buf1 = smem + 12288;    // 4096 floats
    const int m   = blockIdx.x;
    const int tid = threadIdx.x;
    const int bs  = c_in - 1;

    if (tid < c_in) {
        float v = (tid < bs) ? (float)((m >> (bs - 1 - tid)) & 1) : sow[0];
        buf0[tid] = v;             // x[c, l=0]
    }
    __syncthreads();

    float* cur = buf0;
    float* nxt = buf1;
    int in_c = c_in, in_l = 1;
    for (int j = 0; j < n_layers; ++j) {
        const int out_c = in_c * 2, out_l = in_l * 2;
        const float* __restrict__ W    = p.W[j];
        const float* __restrict__ bias = p.b[j];
        const float* __restrict__ gam  = p.g[j];
        const float* __restrict__ bet  = p.bt[j];
        const float* __restrict__ mean = p.mn[j];
        const float* __restrict__ var  = p.vr[j];
        const int total = out_c * out_l;
        for (int e = tid; e < total; e += blockDim.x) {
            const int o   = e / out_l;
            const int pos = e - o * out_l;
            const int l   = pos >> 1;
            const int kk  = pos & 1;
            // out[o, 2l+kk] = sum_c x[c,l] * W[c,o,kk]
            float sum = 0.f;
            for (int c = 0; c < in_c; ++c)
                sum += cur[c * in_l + l] * W[(c * out_c + o) * 2 + kk];
            const float scale = gam[o] / sqrtf(var[o] + 1e-5f);
            float v = (sum + bias[o] - mean[o]) * scale + bet[o];
            nxt[e] = v > 0.f ? v : 0.f;
        }
        __syncthreads();
        float* t = cur; cur = nxt; nxt = t;
        in_c = out_c; in_l = out_l;
    }
    const int K = in_c * in_l;
    for (int e = tid; e < K; e += blockDim.x)
        act[(size_t)m * K + e] = cur[e];
}

// ---------------------------------------------------------------------------
// Phase B: C[nb,4096] = act[nb,K] * linW[4096,K]^T + linB via
// V_WMMA_F32_16X16X4_F32. One wave per 16x16 output tile.
//
// Padding rows (m >= nb) duplicate row 0 (pointer clamp) -- WMMA rows are
// independent, and those rows are never stored, so no zeroing/divergence.
//
// K is consumed in a permuted order so each lane issues contiguous b128
// loads: lo lanes load floats [k0, k0+4), hi lanes [k0+4, k0+8); the .xy /
// .zw halves of each float4 feed two WMMA fragments (k-sets
// {k0,k0+1,k0+4,k0+5} and {k0+2,k0+3,k0+6,k0+7}). A and B use the identical
// mapping, so the K-reduction is exact.
//
// grid = 32 blocks x 256 threads = 256 waves = 256 column tiles (4096 cols).
// ---------------------------------------------------------------------------
__global__ __launch_bounds__(256)
void hnet_branch_gemm(const float* __restrict__ act,
                      const float* __restrict__ linW,
                      const float* __restrict__ linB,
                      float* __restrict__ out,
                      int K, int nb, int origBase) {
    const int lane    = threadIdx.x & 31;
    const int wave    = threadIdx.x >> 5;
    const int tile    = blockIdx.x * 8 + wave;     // 0..255
    const int colBase = tile * 16;
    const int hi      = lane >> 4;                 // lane-half: k offset +4
    const int mn      = lane & 15;                 // M for A-frag, N for B-frag
    const int mClamp  = (mn < nb) ? mn : 0;        // pad rows duplicate row 0

    // Per-lane base pointers, already offset by the lane-half (+4 floats).
    const v4f* __restrict__ aPtr = reinterpret_cast<const v4f*>(
        act + (size_t)mClamp * (size_t)K + hi * 4);
    const v4f* __restrict__ bPtr = reinterpret_cast<const v4f*>(
        linW + (size_t)(colBase + mn) * (size_t)K + hi * 4);

    v8f acc0 = {}, acc1 = {}, acc2 = {}, acc3 = {};

    // K is always a multiple of 32 (4096/8192/12288): 4 b128 A loads,
    // 4 non-temporal b128 B loads, 8 WMMAs on 4 independent acc chains.
    const int nIter = K >> 5;                      // 32 k per iteration
    for (int it = 0; it < nIter; ++it) {
        const int q = it * 8;                      // v4f index (8 floats/step)
        v4f a0 = aPtr[q + 0];
        v4f a1 = aPtr[q + 2];
        v4f a2 = aPtr[q + 4];
        v4f a3 = aPtr[q + 6];
        v4f b0 = __builtin_nontemporal_load(bPtr + q + 0);
        v4f b1 = __builtin_nontemporal_load(bPtr + q + 2);
        v4f b2 = __builtin_nontemporal_load(bPtr + q + 4);
        v4f b3 = __builtin_nontemporal_load(bPtr + q + 6);

        v2f aL, bL;
        aL = (v2f){a0.x, a0.y}; bL = (v2f){b0.x, b0.y};
        acc0 = __builtin_amdgcn_wmma_f32_16x16x4_f32(false, aL, false, bL,
                                                     (short)0, acc0, false, false);
        aL = (v2f){a0.z, a0.w}; bL = (v2f){b0.z, b0.w};
        acc1 = __builtin_amdgcn_wmma_f32_16x16x4_f32(false, aL, false, bL,
                                                     (short)0, acc1, false, false);
        aL = (v2f){a1.x, a1.y}; bL = (v2f){b1.x, b1.y};
        acc2 = __builtin_amdgcn_wmma_f32_16x16x4_f32(false, aL, false, bL,
                                                     (short)0, acc2, false, false);
        aL = (v2f){a1.z, a1.w}; bL = (v2f){b1.z, b1.w};
        acc3 = __builtin_amdgcn_wmma_f32_16x16x4_f32(false, aL, false, bL,
                                                     (short)0, acc3, false, false);
        aL = (v2f){a2.x, a2.y}; bL = (v2f){b2.x, b2.y};
        acc0 = __builtin_amdgcn_wmma_f32_16x16x4_f32(false, aL, false, bL,
                                                     (short)0, acc0, false, false);
        aL = (v2f){a2.z, a2.w}; bL = (v2f){b2.z, b2.w};
        acc1 = __builtin_amdgcn_wmma_f32_16x16x4_f32(false, aL, false, bL,
                                                     (short)0, acc1, false, false);
        aL = (v2f){a3.x, a3.y}; bL = (v2f){b3.x, b3.y};
        acc2 = __builtin_amdgcn_wmma_f32_16x16x4_f32(false, aL, false, bL,
                                                     (short)0, acc2, false, false);
        aL = (v2f){a3.z, a3.w}; bL = (v2f){b3.z, b3.w};
        acc3 = __builtin_amdgcn_wmma_f32_16x16x4_f32(false, aL, false, bL,
                                                     (short)0, acc3, false, false);
    }
    v8f acc = (acc0 + acc1) + (acc2 + acc3);

    const int   col  = colBase + mn;
    const float bias = linB[col];
#pragma unroll
    for (int v = 0; v < 8; ++v) {
        const int mrow = v + hi * 8;               // C/D layout: M = v + 8*hi
        if (mrow < nb) {
            const int orow = d_inv_order[origBase + mrow];
            out[(size_t)orow * 4096 + col] = acc[v] + bias;
        }
    }
}

// ---------------------------------------------------------------------------
extern "C" void kernel_launch(void* const* d_in, const int* in_sizes, int n_in,
                              void* d_out, int out_size, void* d_ws, size_t ws_size,
                              hipStream_t stream) {
    static const int NL[7]   = {5, 6, 6, 6, 6, 6, 6};       // layers per branch
    static const int CIN[7]  = {4, 3, 2, 2, 3, 3, 2};       // base_size + 1
    static const int NB[7]   = {8, 4, 2, 2, 4, 4, 2};       // rows per branch
    static const int KT[7]   = {4096, 12288, 8192, 8192, 12288, 12288, 8192};
    static const int BASE[7] = {0, 8, 12, 14, 16, 20, 24};  // original row base

    // d_in layout: [0]=SoW; then per branch {(W,b,gamma,beta)*NL, linW, linB};
    // then per branch {(mean,var)*NL}.
    int layerIdx[7][6], linWIdx[7], linBIdx[7], statIdx[7][6];
    int cursor = 1;
    for (int i = 0; i < 7; ++i) {
        for (int j = 0; j < NL[i]; ++j) { layerIdx[i][j] = cursor; cursor += 4; }
        linWIdx[i] = cursor++;
        linBIdx[i] = cursor++;
    }
    for (int i = 0; i < 7; ++i)
        for (int j = 0; j < NL[i]; ++j) { statIdx[i][j] = cursor; cursor += 2; }

    const float* sow = (const float*)d_in[0];
    float* ws  = (float*)d_ws;
    float* out = (float*)d_out;

    size_t actOff[7];
    size_t off = 0;
    for (int i = 0; i < 7; ++i) { actOff[i] = off; off += (size_t)NB[i] * KT[i]; }

    // Phase A: 7 tiny pipeline kernels (26 workgroups total).
    for (int i = 0; i < 7; ++i) {
        BranchParams bp{};
        for (int j = 0; j < NL[i]; ++j) {
            const int b = layerIdx[i][j];
            bp.W[j]  = (const float*)d_in[b + 0];
            bp.b[j]  = (const float*)d_in[b + 1];
            bp.g[j]  = (const float*)d_in[b + 2];
            bp.bt[j] = (const float*)d_in[b + 3];
            bp.mn[j] = (const float*)d_in[statIdx[i][j] + 0];
            bp.vr[j] = (const float*)d_in[statIdx[i][j] + 1];
        }
        hipLaunchKernelGGL(hnet_branch_forward, dim3(NB[i]), dim3(256),
                           65536 /* (12288+4096) floats */, stream,
                           bp, sow, ws + actOff[i], CIN[i], NL[i]);
    }

    // Phase B: 7 weight-streaming WMMA GEMMs (bandwidth-bound part).
    for (int i = 0; i < 7; ++i) {
        hipLaunchKernelGGL(hnet_branch_gemm, dim3(32), dim3(256), 0, stream,
                           ws + actOff[i],
                           (const float*)d_in[linWIdx[i]],
                           (const float*)d_in[linBIdx[i]],
                           out, KT[i], NB[i], BASE[i]);
    }
}